// SimpleLSTM_16320875725179
// MI455X (gfx1250) — compile-verified
//
#include <hip/hip_runtime.h>
#include <math.h>

#define N_T   16384
#define DIN   1024
#define HDIM  1024
#define G4    4096   // 4*HDIM
#define NWG   512    // persistent workgroups in the scan

typedef float v2f __attribute__((ext_vector_type(2)));
typedef float v8f __attribute__((ext_vector_type(8)));

// ---------------------------------------------------------------------------
// Phase 1: x_gates[N_T,4H] = xs @ W_ih^T + (b_ih + b_hh), f32 WMMA 16x16x4.
// Each wave computes a 16(M) x 64(N) strip: 4 accumulator tiles reuse one A
// fragment per k-step. Fragment layouts follow the CDNA5 ISA 32-bit maps:
//   A 16x4 : lanes 0-15 hold K=0,1 ; lanes 16-31 hold K=2,3 (M = lane&15)
//   B 4x16 : lanes 0-15 hold K=0,1 ; lanes 16-31 hold K=2,3 (N = lane&15)
//   C 16x16: VGPR v -> M=v (lanes 0-15), M=v+8 (lanes 16-31), N = lane&15
// ---------------------------------------------------------------------------
__global__ __launch_bounds__(256) void xgates_gemm_wmma(
    const float* __restrict__ xs, const float* __restrict__ Wih,
    const float* __restrict__ bih, const float* __restrict__ bhh,
    float* __restrict__ xg)
{
  const int wv     = threadIdx.x >> 5;
  const int lane   = threadIdx.x & 31;
  const int waveId = blockIdx.x * 8 + wv;
  const int m0 = (waveId >> 6) << 4;   // timestep-tile origin (1024 tiles)
  const int n0 = (waveId & 63) << 6;   // gate-column group origin (64 cols)

  const int half = lane >> 4;          // 0 -> K={0,1}, 1 -> K={2,3}
  const int l15  = lane & 15;

  v8f acc[4];
  #pragma unroll
  for (int j = 0; j < 4; ++j) acc[j] = (v8f)0.f;

  const float* arow = xs + (size_t)(m0 + l15) * DIN + half * 2;

  #pragma unroll 4
  for (int k0 = 0; k0 < DIN; k0 += 4) {
    v2f a = *(const v2f*)(arow + k0);
    #pragma unroll
    for (int j = 0; j < 4; ++j) {
      const int col = n0 + j * 16 + l15;              // W_ih row == gate col
      v2f b = *(const v2f*)(Wih + (size_t)col * DIN + k0 + half * 2);
      acc[j] = __builtin_amdgcn_wmma_f32_16x16x4_f32(
          /*neg_a=*/false, a, /*neg_b=*/false, b,
          /*c_mod=*/(short)0, acc[j], /*reuse_a=*/false, /*reuse_b=*/false);
    }
  }

  #pragma unroll
  for (int j = 0; j < 4; ++j) {
    const int col  = n0 + j * 16 + l15;
    const float bs = bih[col] + bhh[col];
    #pragma unroll
    for (int v = 0; v < 8; ++v) {
      const int row = m0 + v + half * 8;
      xg[(size_t)row * G4 + col] = acc[j][v] + bs;
    }
  }
}

// ---------------------------------------------------------------------------
// Workspace init: zero barrier words and both h ping-pong buffers (h0 = 0).
// ---------------------------------------------------------------------------
__global__ void scan_init(unsigned* bar, float* hbuf) {
  unsigned i = blockIdx.x * blockDim.x + threadIdx.x;
  if (i < 2) bar[i] = 0u;
  for (unsigned j = i; j < 2 * HDIM; j += gridDim.x * blockDim.x) hbuf[j] = 0.f;
}

// ---------------------------------------------------------------------------
// Device-wide sense/generation barrier (one designated lane per WG), with
// polite s_sleep spinning.
// ---------------------------------------------------------------------------
__device__ inline void grid_barrier(unsigned* cnt, unsigned* gen) {
  __syncthreads();
  if (threadIdx.x == 0) {
    __threadfence();
    unsigned g = __hip_atomic_load(gen, __ATOMIC_RELAXED, __HIP_MEMORY_SCOPE_AGENT);
    if (atomicAdd(cnt, 1u) == NWG - 1) {
      atomicExch(cnt, 0u);
      __hip_atomic_store(gen, g + 1u, __ATOMIC_RELEASE, __HIP_MEMORY_SCOPE_AGENT);
    } else {
      while (__hip_atomic_load(gen, __ATOMIC_ACQUIRE, __HIP_MEMORY_SCOPE_AGENT) == g)
        __builtin_amdgcn_s_sleep(1);
    }
    __threadfence();
  }
  __syncthreads();
}

// ---------------------------------------------------------------------------
// Phase 2: sequential LSTM scan. W_hh is fully register-resident:
// 512 WGs x 8 waves x 32 lanes x 32 VGPRs = 4M floats = all of W_hh.
// Wave (wg, wv): unit = 2*wg + wv/4, gate = wv&3, row = gate*H + unit, so all
// four gates of a hidden unit live in one workgroup. Per step: stage h into
// LDS, register-dot + wave32 shuffle reduce, gate math, one grid barrier.
// ---------------------------------------------------------------------------
__global__ __launch_bounds__(256) void lstm_scan(
    const float* __restrict__ Whh, const float* __restrict__ xg,
    float* __restrict__ hs, float* hbuf, unsigned* bar)
{
  __shared__ float s_h[HDIM];
  __shared__ float s_g[8];

  const int wv   = threadIdx.x >> 5;
  const int lane = threadIdx.x & 31;
  const int unit = blockIdx.x * 2 + (wv >> 2);
  const int gate = wv & 3;
  const int row  = gate * HDIM + unit;

  // Preload this wave's W_hh row: lane holds k = lane + 32*q (coalesced).
  float w[32];
  #pragma unroll
  for (int q = 0; q < 32; ++q)
    w[q] = Whh[(size_t)row * HDIM + lane + 32 * q];

  float c = 0.f;
  const bool writer = (lane == 0) && (gate == 0);

  for (int t = 0; t < N_T; ++t) {
    const float* hsrc = hbuf + ((t & 1) ? HDIM : 0);
    float*       hdst = hbuf + ((t & 1) ? 0 : HDIM);

    for (int i = threadIdx.x; i < HDIM; i += 256) s_h[i] = hsrc[i];
    __syncthreads();

    float p = 0.f;
    #pragma unroll
    for (int q = 0; q < 32; ++q) p = fmaf(w[q], s_h[lane + 32 * q], p);
    #pragma unroll
    for (int off = 16; off > 0; off >>= 1) p += __shfl_down(p, off, 32);

    if (lane == 0) s_g[wv] = p + xg[(size_t)t * G4 + row];
    __syncthreads();

    if (writer) {
      const int b = wv;  // 0 or 4
      const float gi = s_g[b + 0], gf = s_g[b + 1];
      const float gg = s_g[b + 2], go = s_g[b + 3];
      const float i_ = 1.f / (1.f + __expf(-gi));
      const float f_ = 1.f / (1.f + __expf(-gf));
      const float g_ = tanhf(gg);
      const float o_ = 1.f / (1.f + __expf(-go));
      c = fmaf(f_, c, i_ * g_);
      const float h = o_ * tanhf(c);
      hs[(size_t)t * HDIM + unit] = h;
      hdst[unit] = h;
      __threadfence();
    }
    grid_barrier(bar, bar + 1);
  }
}

// ---------------------------------------------------------------------------
extern "C" void kernel_launch(void* const* d_in, const int* in_sizes, int n_in,
                              void* d_out, int out_size, void* d_ws, size_t ws_size,
                              hipStream_t stream) {
  const float* xs  = (const float*)d_in[0];   // [16384,1024]
  const float* Wih = (const float*)d_in[1];   // [4096,1024]
  const float* Whh = (const float*)d_in[2];   // [4096,1024]
  const float* bih = (const float*)d_in[3];   // [4096]
  const float* bhh = (const float*)d_in[4];   // [4096]
  float* hs = (float*)d_out;                  // [16384,1024]

  char* ws      = (char*)d_ws;
  unsigned* bar = (unsigned*)ws;              // 2 words: count, generation
  float* hbuf   = (float*)(ws + 256);         // ping-pong h: 2*1024 floats
  float* xg     = (float*)(ws + 16384);       // x_gates: 16384*4096 floats

  scan_init<<<2, 1024, 0, stream>>>(bar, hbuf);
  xgates_gemm_wmma<<<8192, 256, 0, stream>>>(xs, Wih, bih, bhh, xg);
  lstm_scan<<<NWG, 256, 0, stream>>>(Whh, xg, hs, hbuf, bar);
}